// ImprovedSpatialEncoder_5068061409445
// MI455X (gfx1250) — compile-verified
//
#include <hip/hip_runtime.h>

typedef __attribute__((ext_vector_type(2))) float v2f;
typedef __attribute__((ext_vector_type(8))) float v8f;

#define HDIM 128

// ---------------- degree / normalization ----------------

__global__ void k_deg_init(float* __restrict__ deg, int n) {
    int i = blockIdx.x * 256 + threadIdx.x;
    if (i < n) deg[i] = 1.0f;  // self-loop contribution
}

__global__ void k_deg_accum(const int* __restrict__ dst, float* __restrict__ deg, int e) {
    int i = blockIdx.x * 256 + threadIdx.x;
    if (i < e) atomicAdd(&deg[dst[i]], 1.0f);
}

__global__ void k_deg_rsqrt(float* __restrict__ deg, int n) {
    int i = blockIdx.x * 256 + threadIdx.x;
    if (i < n) {
        float d = deg[i];
        deg[i] = (d > 0.0f) ? rsqrtf(d) : 0.0f;
    }
}

// ---------------- fused GEMM via V_WMMA_F32_16X16X4_F32 ----------------
// AMODE (compile-time): 0: A = X     1: A = relu(X)     2: A = relu(X) + Res
// HAS_AGG (compile-time):
//   false: t_out = A@W + bias
//   true : t_out = A@W ;  agg_out = bias + (A@W)*dinv^2   (bias + self-loop msg)
// One wave per 16x16 output tile; 8 waves/block share the same 16 A-rows.
template <int AMODE, bool HAS_AGG>
__global__ void __launch_bounds__(256)
k_gemm_wmma(const float* __restrict__ X, const float* __restrict__ Res,
            const float* __restrict__ W, const float* __restrict__ bias,
            const float* __restrict__ dinv,
            float* __restrict__ t_out, float* __restrict__ agg_out, int n) {
    const int wave = threadIdx.x >> 5;   // 0..7 -> column tile
    const int lane = threadIdx.x & 31;
    const int m    = lane & 15;          // A row / B col / D col within tile
    const int hi   = lane >> 4;          // 0/1: K split across lane halves

    const int row0 = blockIdx.x * 16;
    int arow = row0 + m;
    if (arow >= n) arow = n - 1;         // clamp: EXEC must stay all-1s for WMMA
    const int col = wave * 16 + m;

    const float* __restrict__ xrow = X + (size_t)arow * HDIM;
    const float* __restrict__ rrow = Res + (size_t)arow * HDIM;

    v8f c = {};
    #pragma unroll
    for (int k = 0; k < HDIM; k += 4) {
        const int ka = k + 2 * hi;       // lanes 0-15: K=k,k+1 ; lanes 16-31: K=k+2,k+3
        float ax = xrow[ka];
        float ay = xrow[ka + 1];
        if (AMODE != 0) {                // fused ReLU of previous layer's aggregate
            ax = fmaxf(ax, 0.0f);
            ay = fmaxf(ay, 0.0f);
        }
        if (AMODE == 2) {                // fused residual add (layer-0 output)
            ax += rrow[ka];
            ay += rrow[ka + 1];
        }
        v2f a; a.x = ax; a.y = ay;
        v2f b;
        b.x = W[(size_t)ka * HDIM + col];
        b.y = W[(size_t)(ka + 1) * HDIM + col];
        // (neg_a, A, neg_b, B, c_mod, C, reuse_a, reuse_b)
        c = __builtin_amdgcn_wmma_f32_16x16x4_f32(false, a, false, b, (short)0, c,
                                                  false, false);
    }

    const float badd = bias[col];
    if (HAS_AGG) {
        #pragma unroll
        for (int v = 0; v < 8; ++v) {
            const int r = row0 + v + 8 * hi;  // C/D layout: VGPR v + lane half -> row
            if (r < n) {
                const float cv = c[v];
                const size_t off = (size_t)r * HDIM + col;
                t_out[off] = cv;                       // gather source for scatter
                const float di = dinv[r];
                agg_out[off] = badd + cv * di * di;    // bias + self-loop message
            }
        }
    } else {
        #pragma unroll
        for (int v = 0; v < 8; ++v) {
            const int r = row0 + v + 8 * hi;
            if (r < n) t_out[(size_t)r * HDIM + col] = c[v] + badd;
        }
    }
}

// ---------------- edge scatter ----------------
// out[dst[e],:] += t[src[e],:] * dinv[src]*dinv[dst]   (32 threads x float4 / edge)
__global__ void k_scatter(const float* __restrict__ t, const int* __restrict__ src,
                          const int* __restrict__ dst, const float* __restrict__ dinv,
                          float* __restrict__ out, int e) {
    int idx = blockIdx.x * 256 + threadIdx.x;
    int ei = idx >> 5;
    int j  = (idx & 31) * 4;
    if (ei >= e) return;
    int s = src[ei];
    int d = dst[ei];
    float w = dinv[s] * dinv[d];
    const float4 v = *(const float4*)(t + (size_t)s * HDIM + j);
    float* o = out + (size_t)d * HDIM + j;
    atomicAdd(o + 0, v.x * w);
    atomicAdd(o + 1, v.y * w);
    atomicAdd(o + 2, v.z * w);
    atomicAdd(o + 3, v.w * w);
}

// ---------------- final activation ----------------
__global__ void k_relu(const float* __restrict__ agg, float* __restrict__ out, int n) {
    int idx = blockIdx.x * 256 + threadIdx.x;
    int i = idx >> 5;
    int j = (idx & 31) * 4;
    if (i >= n) return;
    size_t off = (size_t)i * HDIM + j;
    float4 a = *(const float4*)(agg + off);
    a.x = fmaxf(a.x, 0.0f);
    a.y = fmaxf(a.y, 0.0f);
    a.z = fmaxf(a.z, 0.0f);
    a.w = fmaxf(a.w, 0.0f);
    *(float4*)(out + off) = a;
}

// ---------------- driver ----------------

extern "C" void kernel_launch(void* const* d_in, const int* in_sizes, int n_in,
                              void* d_out, int out_size, void* d_ws, size_t ws_size,
                              hipStream_t stream) {
    const float* x    = (const float*)d_in[0];
    const int*   edge = (const int*)  d_in[1];
    const float* W0   = (const float*)d_in[2];
    const float* b0   = (const float*)d_in[3];
    const float* W1   = (const float*)d_in[4];
    const float* b1   = (const float*)d_in[5];
    const float* W2   = (const float*)d_in[6];
    const float* b2   = (const float*)d_in[7];
    const float* Wr   = (const float*)d_in[8];
    const float* br   = (const float*)d_in[9];

    const int n = in_sizes[0] / HDIM;
    const int e = in_sizes[1] / 2;
    const int* srcp = edge;       // edge_index[0]
    const int* dstp = edge + e;   // edge_index[1]

    float* ws   = (float*)d_ws;
    float* dinv = ws;
    const size_t nh = (size_t)n * HDIM;
    float* bufA = ws + (((size_t)n + 3) & ~(size_t)3);  // keep 16B alignment
    float* bufB = bufA + nh;
    float* bufC = bufB + nh;
    float* out  = (float*)d_out;

    const dim3 blk(256);
    const int gN  = (n + 255) / 256;
    const int gE  = (e + 255) / 256;
    const int gNH = (int)(((size_t)n * 32 + 255) / 256);
    const int gEH = (int)(((size_t)e * 32 + 255) / 256);
    const int gT  = (n + 15) / 16;

    // D^{-1/2}
    k_deg_init <<<gN, blk, 0, stream>>>(dinv, n);
    k_deg_accum<<<gE, blk, 0, stream>>>(dstp, dinv, e);
    k_deg_rsqrt<<<gN, blk, 0, stream>>>(dinv, n);

    // residual = x @ Wr + br  -> bufC
    k_gemm_wmma<0, false><<<gT, blk, 0, stream>>>(x, x, Wr, br, dinv, bufC, nullptr, n);

    // ---- layer 0 ----  t0 -> bufA, agg0 (bias+self) -> bufB, scatter into bufB
    k_gemm_wmma<0, true><<<gT, blk, 0, stream>>>(x, x, W0, b0, dinv, bufA, bufB, n);
    k_scatter <<<gEH, blk, 0, stream>>>(bufA, srcp, dstp, dinv, bufB, e);

    // ---- layer 1 ----  A = relu(bufB) + bufC ; t1 -> bufA, agg1 -> d_out
    k_gemm_wmma<2, true><<<gT, blk, 0, stream>>>(bufB, bufC, W1, b1, dinv, bufA, out, n);
    k_scatter <<<gEH, blk, 0, stream>>>(bufA, srcp, dstp, dinv, out, e);

    // ---- layer 2 ----  A = relu(d_out) ; t2 -> bufB, agg2 -> bufC
    k_gemm_wmma<1, true><<<gT, blk, 0, stream>>>(out, out, W2, b2, dinv, bufB, bufC, n);
    k_scatter <<<gEH, blk, 0, stream>>>(bufB, srcp, dstp, dinv, bufC, e);

    // final: d_out = relu(agg2)
    k_relu    <<<gNH, blk, 0, stream>>>(bufC, out, n);
}